// SparseAttentionAggregator_20538533609704
// MI455X (gfx1250) — compile-verified
//
#include <hip/hip_runtime.h>
#include <hip/hip_bf16.h>

typedef float v2f __attribute__((ext_vector_type(2)));
typedef float v8f __attribute__((ext_vector_type(8)));

#define S_F   32
#define P_T   196
#define NH    16
#define HD    64
#define CDIM  1024
#define NTOK  6272          // S_F * P_T
#define MKV   1960
#define MKPAD 1984          // 31 * 64
#define KCH   64            // key chunk
#define ATT_WAVES 13        // 13*16 = 208 >= 196 query rows
#define KSTR  68            // LDS row stride (floats) for K chunk (transposed reads)
#define PSTR  68            // LDS row stride for P scratch
#define LN_EPS 1e-6f

static __device__ __forceinline__ v8f wmma_f32(v2f a, v2f b, v8f c) {
    // D = A(16x4,f32) * B(4x16,f32) + C(16x16,f32)
    return __builtin_amdgcn_wmma_f32_16x16x4_f32(false, a, false, b, (short)0, c,
                                                 false, false);
}

static __device__ __forceinline__ float half16_max(float v) {
    v = fmaxf(v, __shfl_xor(v, 1, 32));
    v = fmaxf(v, __shfl_xor(v, 2, 32));
    v = fmaxf(v, __shfl_xor(v, 4, 32));
    v = fmaxf(v, __shfl_xor(v, 8, 32));
    return v;
}
static __device__ __forceinline__ float half16_sum(float v) {
    v += __shfl_xor(v, 1, 32);
    v += __shfl_xor(v, 2, 32);
    v += __shfl_xor(v, 4, 32);
    v += __shfl_xor(v, 8, 32);
    return v;
}

// ---------------------------------------------------------------------------
// Generic fp32 WMMA GEMM:  C[M,N] = A[M,K] @ B[K,N] + bias[N]
// Requires M%128==0, N%128==0, K%16==0. 256 threads (8 waves), 128x128 tile.
// B tile staged K-pair-interleaved: element (k,c) at Bs[(k>>1)*256 + c*2 + (k&1)]
// so every B fragment is a single contiguous ds_load_b64. B fragments are
// double-buffered in registers (loads issued one k-step ahead).
// ---------------------------------------------------------------------------
#define GA_STR 24   // A-tile LDS row stride (floats): 16B aligned, low conflicts
__global__ __launch_bounds__(256)
void gemm_f32_wmma(const float* __restrict__ A, const float* __restrict__ B,
                   const float* __restrict__ bias, float* __restrict__ C,
                   int M, int N, int K) {
    __shared__ float As[128 * GA_STR];
    __shared__ float Bs[8 * 256];        // 8 k-pairs x 128 cols x 2
    const int tid  = threadIdx.x;
    const int wave = tid >> 5;
    const int lane = tid & 31;
    const int l16  = lane & 15;
    const int lh   = lane >> 4;          // 0 or 1 (K-half / M-half selector)
    const int mbase = blockIdx.x * 128;
    const int nbase = blockIdx.y * 128;

    v8f acc[8];
#pragma unroll
    for (int i = 0; i < 8; ++i) {
        v8f z = {0.f,0.f,0.f,0.f,0.f,0.f,0.f,0.f};
        acc[i] = z;
    }

    for (int kb = 0; kb < K; kb += 16) {
        // --- stage A tile (128 rows x 16 cols), row-major, pad 24 ---
#pragma unroll
        for (int t = 0; t < 2; ++t) {
            int i  = tid + t * 256;      // 0..511 float4 slots
            int r  = i >> 2;
            int c4 = i & 3;
            float4 v = *(const float4*)(A + (size_t)(mbase + r) * K + kb + c4 * 4);
            *(float4*)(&As[r * GA_STR + c4 * 4]) = v;
        }
        // --- stage B tile (16 rows x 128 cols) K-pair interleaved ---
#pragma unroll
        for (int t = 0; t < 2; ++t) {
            int i  = tid + t * 256;      // 0..511
            int kp = i >> 6;             // k-pair 0..7
            int c  = (i & 63) * 2;       // col 0,2,..,126
            float2 lo = *(const float2*)(B + (size_t)(kb + 2 * kp    ) * N + nbase + c);
            float2 hi = *(const float2*)(B + (size_t)(kb + 2 * kp + 1) * N + nbase + c);
            float4 v = make_float4(lo.x, hi.x, lo.y, hi.y);
            *(float4*)(&Bs[kp * 256 + c * 2]) = v;
        }
        __syncthreads();

        const int arow = wave * 16 + l16;
        // preload all 4 A fragments (contiguous pairs -> ds_load_2addr_b64 x2)
        v2f a[4];
#pragma unroll
        for (int ks = 0; ks < 4; ++ks)
            a[ks] = *(const v2f*)(&As[arow * GA_STR + ks * 4 + lh * 2]);

        // double-buffered B fragments: issue loads for ks+1 before wmmas of ks
        v2f bb[2][8];
#pragma unroll
        for (int nt = 0; nt < 8; ++nt)
            bb[0][nt] = *(const v2f*)(&Bs[lh * 256 + (nt * 16 + l16) * 2]);
#pragma unroll
        for (int ks = 0; ks < 4; ++ks) {
            if (ks < 3) {
                int kpn = 2 * (ks + 1) + lh;
#pragma unroll
                for (int nt = 0; nt < 8; ++nt)
                    bb[(ks + 1) & 1][nt] =
                        *(const v2f*)(&Bs[kpn * 256 + (nt * 16 + l16) * 2]);
            }
#pragma unroll
            for (int nt = 0; nt < 8; ++nt)
                acc[nt] = wmma_f32(a[ks], bb[ks & 1][nt], acc[nt]);
        }
        __syncthreads();
    }

    // epilogue: C layout -> row = r + 8*lh, col = nt*16 + l16
#pragma unroll
    for (int nt = 0; nt < 8; ++nt) {
        int col = nbase + nt * 16 + l16;
        float bv = bias[col];
#pragma unroll
        for (int r = 0; r < 8; ++r) {
            int row = mbase + wave * 16 + r + lh * 8;
            C[(size_t)row * N + col] = acc[nt][r] + bv;
        }
    }
}

// ---------------------------------------------------------------------------
// LayerNorm(q,k over D=64) + layout reorder.
//   q -> [H, NTOK, D]   (per (h,frame): contiguous [P, D] tile)
//   k,v -> [tok, H, D]  (gather-friendly 256B rows)
// One thread per (token, head).
// ---------------------------------------------------------------------------
__global__ __launch_bounds__(256)
void ln_reorder(const float* __restrict__ qkv,
                float* __restrict__ qw, float* __restrict__ kw, float* __restrict__ vw,
                const float* __restrict__ qg, const float* __restrict__ qb,
                const float* __restrict__ kg, const float* __restrict__ kb) {
    int idx = blockIdx.x * blockDim.x + threadIdx.x;
    if (idx >= NTOK * NH) return;
    int n = idx >> 4;
    int h = idx & 15;
    const float* base = qkv + (size_t)n * (3 * CDIM);

    // q: layernorm then [H, NTOK, D]
    {
        const float* x = base + h * HD;
        float mu = 0.f;
#pragma unroll
        for (int d = 0; d < HD; ++d) mu += x[d];
        mu *= (1.f / HD);
        float var = 0.f;
#pragma unroll
        for (int d = 0; d < HD; ++d) { float t = x[d] - mu; var += t * t; }
        var *= (1.f / HD);
        float rs = rsqrtf(var + LN_EPS);
        float* dst = qw + ((size_t)h * NTOK + n) * HD;
#pragma unroll
        for (int d = 0; d < HD; ++d) dst[d] = (x[d] - mu) * rs * qg[d] + qb[d];
    }
    // k: layernorm then [tok, H, D]
    {
        const float* x = base + CDIM + h * HD;
        float mu = 0.f;
#pragma unroll
        for (int d = 0; d < HD; ++d) mu += x[d];
        mu *= (1.f / HD);
        float var = 0.f;
#pragma unroll
        for (int d = 0; d < HD; ++d) { float t = x[d] - mu; var += t * t; }
        var *= (1.f / HD);
        float rs = rsqrtf(var + LN_EPS);
        float* dst = kw + ((size_t)n * NH + h) * HD;
#pragma unroll
        for (int d = 0; d < HD; ++d) dst[d] = (x[d] - mu) * rs * kg[d] + kb[d];
    }
    // v: copy to [tok, H, D]
    {
        const float* x = base + 2 * CDIM + h * HD;
        float* dst = vw + ((size_t)n * NH + h) * HD;
#pragma unroll
        for (int d = 0; d < HD; ++d) dst[d] = x[d];
    }
}

// ---------------------------------------------------------------------------
// Sparse gathered attention, flash-style online softmax, all WMMA f32.
// grid = (S_F frames, NH heads); block = 13 waves; wave w owns query rows
// [16w, 16w+16). Keys streamed in 64-wide chunks through LDS.
// V chunk staged key-pair-interleaved: (key,d) at Vp[(key>>1)*128 + d*2 + (key&1)]
// so every V fragment is one contiguous ds_load_b64. Fragment streams are
// double-buffered in registers for DS-latency overlap.
// Output written as [NTOK, H*D] ready for the projection GEMM.
// ---------------------------------------------------------------------------
__global__ __launch_bounds__(ATT_WAVES * 32)
void attn_sparse(const float* __restrict__ qw, const float* __restrict__ kw,
                 const float* __restrict__ vw, const int* __restrict__ gidx,
                 const float* __restrict__ abias, float* __restrict__ outw) {
    __shared__ float Kc[KCH * KSTR];
    __shared__ float Vp[(KCH / 2) * 128];          // 32 key-pairs x 64 dims x 2
    __shared__ float Ps[ATT_WAVES * 16 * PSTR];

    const int s    = blockIdx.x;
    const int h    = blockIdx.y;
    const int tid  = threadIdx.x;
    const int wave = tid >> 5;
    const int lane = tid & 31;
    const int l16  = lane & 15;
    const int lh   = lane >> 4;

    const int* gi   = gidx + s * MKV;
    const float* br = abias + s * MKV;

    // Load Q fragments (prescaled by 1/sqrt(D)=0.125). A-frag: lane -> row l16,
    // ks-th fragment holds K dims 4*ks + 2*lh + {0,1}.
    int p  = wave * 16 + l16;
    int pc = p < P_T ? p : (P_T - 1);
    const float* qrow = qw + ((size_t)h * NTOK + (size_t)s * P_T + pc) * HD;
    v2f qa[16];
#pragma unroll
    for (int ks = 0; ks < 16; ++ks) {
        int k0 = ks * 4 + lh * 2;
        v2f t = *(const v2f*)(qrow + k0);
        qa[ks].x = t.x * 0.125f;
        qa[ks].y = t.y * 0.125f;
    }

    v8f o[4];
#pragma unroll
    for (int i = 0; i < 4; ++i) {
        v8f z = {0.f,0.f,0.f,0.f,0.f,0.f,0.f,0.f};
        o[i] = z;
    }
    float mrow[8], lrow[8];
#pragma unroll
    for (int r = 0; r < 8; ++r) { mrow[r] = -1e30f; lrow[r] = 0.f; }

    float* pw = Ps + wave * 16 * PSTR;

    for (int base = 0; base < MKPAD; base += KCH) {
        __syncthreads();   // protect Kc/Vp from previous iteration's readers
        // cooperative gather: K rows (row-major, padded stride)
        for (int i = tid; i < KCH * (HD / 2); i += ATT_WAVES * 32) {
            int key = i >> 5;              // 32 float2 per 64-float row
            int c2  = (i & 31) * 2;
            int kg2 = base + key;
            int tok = gi[kg2 < MKV ? kg2 : 0];
            *(float2*)(&Kc[key * KSTR + c2]) =
                *(const float2*)(kw + ((size_t)tok * NH + h) * HD + c2);
        }
        // cooperative gather: V rows, key-pair interleaved
        for (int i = tid; i < (KCH / 2) * (HD / 2); i += ATT_WAVES * 32) {
            int kp = i >> 5;               // key pair 0..31
            int c  = (i & 31) * 2;         // dim 0,2,..,62
            int kg0 = base + 2 * kp;
            int kg1 = kg0 + 1;
            int t0 = gi[kg0 < MKV ? kg0 : 0];
            int t1 = gi[kg1 < MKV ? kg1 : 0];
            float2 lo = *(const float2*)(vw + ((size_t)t0 * NH + h) * HD + c);
            float2 hi = *(const float2*)(vw + ((size_t)t1 * NH + h) * HD + c);
            float4 v = make_float4(lo.x, hi.x, lo.y, hi.y);
            *(float4*)(&Vp[kp * 128 + c * 2]) = v;
        }
        __syncthreads();

        // ---- scores: S = (Q*scale) @ K^T   [16 x 64] in 4 N-tiles ----
        v8f st[4];
#pragma unroll
        for (int nt = 0; nt < 4; ++nt) {
            v8f z = {0.f,0.f,0.f,0.f,0.f,0.f,0.f,0.f};
            st[nt] = z;
        }
        {
            v2f kb2[2][4];
#pragma unroll
            for (int nt = 0; nt < 4; ++nt)
                kb2[0][nt] = *(const v2f*)(&Kc[(nt * 16 + l16) * KSTR + lh * 2]);
#pragma unroll
            for (int ks = 0; ks < 16; ++ks) {
                if (ks < 15) {
                    int k0n = (ks + 1) * 4 + lh * 2;
#pragma unroll
                    for (int nt = 0; nt < 4; ++nt)
                        kb2[(ks + 1) & 1][nt] =
                            *(const v2f*)(&Kc[(nt * 16 + l16) * KSTR + k0n]);
                }
#pragma unroll
                for (int nt = 0; nt < 4; ++nt)
                    st[nt] = wmma_f32(qa[ks], kb2[ks & 1][nt], st[nt]);
            }
        }

        // ---- bias ----
#pragma unroll
        for (int nt = 0; nt < 4; ++nt) {
            int kg2 = base + nt * 16 + l16;
            float bv = (kg2 < MKV) ? br[kg2] : -1e9f;
#pragma unroll
            for (int r = 0; r < 8; ++r) st[nt][r] += bv;
        }

        // ---- online softmax update ----
        float ef[8];
#pragma unroll
        for (int r = 0; r < 8; ++r) {
            float mx = fmaxf(fmaxf(st[0][r], st[1][r]), fmaxf(st[2][r], st[3][r]));
            mx = half16_max(mx);
            float mn = fmaxf(mrow[r], mx);
            ef[r] = __expf(mrow[r] - mn);
            mrow[r] = mn;
        }
#pragma unroll
        for (int nt = 0; nt < 4; ++nt)
#pragma unroll
            for (int r = 0; r < 8; ++r)
                st[nt][r] = __expf(st[nt][r] - mrow[r]);
#pragma unroll
        for (int r = 0; r < 8; ++r) {
            float sm = st[0][r] + st[1][r] + st[2][r] + st[3][r];
            sm = half16_sum(sm);
            lrow[r] = lrow[r] * ef[r] + sm;
        }
#pragma unroll
        for (int nt = 0; nt < 4; ++nt)
#pragma unroll
            for (int r = 0; r < 8; ++r)
                o[nt][r] *= ef[r];

        // ---- transpose P through per-wave LDS into A-fragment layout ----
#pragma unroll
        for (int nt = 0; nt < 4; ++nt)
#pragma unroll
            for (int r = 0; r < 8; ++r)
                pw[(r + 8 * lh) * PSTR + nt * 16 + l16] = st[nt][r];
        // same-wave LDS ops are in-order; compiler inserts DScnt waits

        // ---- O += P @ V   (contraction over 64 keys) ----
        {
            v2f pa[2], vb[2][4];
            pa[0] = *(const v2f*)(&pw[l16 * PSTR + lh * 2]);
#pragma unroll
            for (int nt = 0; nt < 4; ++nt)
                vb[0][nt] = *(const v2f*)(&Vp[lh * 128 + (nt * 16 + l16) * 2]);
#pragma unroll
            for (int ks = 0; ks < 16; ++ks) {
                if (ks < 15) {
                    int k0n = (ks + 1) * 4 + lh * 2;
                    int kpn = 2 * (ks + 1) + lh;
                    pa[(ks + 1) & 1] = *(const v2f*)(&pw[l16 * PSTR + k0n]);
#pragma unroll
                    for (int nt = 0; nt < 4; ++nt)
                        vb[(ks + 1) & 1][nt] =
                            *(const v2f*)(&Vp[kpn * 128 + (nt * 16 + l16) * 2]);
                }
#pragma unroll
                for (int nt = 0; nt < 4; ++nt)
                    o[nt] = wmma_f32(pa[ks & 1], vb[ks & 1][nt], o[nt]);
            }
        }
    }

    // ---- normalize + store as [NTOK, H*D] ----
    float inv[8];
#pragma unroll
    for (int r = 0; r < 8; ++r) inv[r] = 1.f / lrow[r];
#pragma unroll
    for (int nt = 0; nt < 4; ++nt) {
#pragma unroll
        for (int r = 0; r < 8; ++r) {
            int pr = wave * 16 + r + 8 * lh;
            if (pr < P_T) {
                size_t n = (size_t)s * P_T + pr;
                outw[(n * NH + h) * HD + nt * 16 + l16] = o[nt][r] * inv[r];
            }
        }
    }
}

// ---------------------------------------------------------------------------
extern "C" void kernel_launch(void* const* d_in, const int* in_sizes, int n_in,
                              void* d_out, int out_size, void* d_ws, size_t ws_size,
                              hipStream_t stream) {
    (void)in_sizes; (void)n_in; (void)out_size; (void)ws_size;
    const float* x    = (const float*)d_in[0];
    const float* Wqkv = (const float*)d_in[1];
    const float* bqkv = (const float*)d_in[2];
    const float* qg   = (const float*)d_in[3];
    const float* qb   = (const float*)d_in[4];
    const float* kg   = (const float*)d_in[5];
    const float* kb   = (const float*)d_in[6];
    const float* Wp   = (const float*)d_in[7];
    const float* bp   = (const float*)d_in[8];
    const int*   gix  = (const int*)d_in[9];
    const float* ab   = (const float*)d_in[10];
    float* out = (float*)d_out;

    float* w = (float*)d_ws;
    float* qkv_buf = w;                                       // NTOK*3072
    float* q_ws    = w + (size_t)NTOK * 3 * CDIM;             // NTOK*1024
    float* k_ws    = q_ws + (size_t)NTOK * CDIM;
    float* v_ws    = k_ws + (size_t)NTOK * CDIM;
    float* attn_o  = w;   // aliases qkv_buf (dead after ln_reorder)

    // 1) QKV projection
    dim3 g1(NTOK / 128, (3 * CDIM) / 128);
    gemm_f32_wmma<<<g1, 256, 0, stream>>>(x, Wqkv, bqkv, qkv_buf,
                                          NTOK, 3 * CDIM, CDIM);
    // 2) q/k layernorm + layout reorder
    ln_reorder<<<(NTOK * NH + 255) / 256, 256, 0, stream>>>(
        qkv_buf, q_ws, k_ws, v_ws, qg, qb, kg, kb);
    // 3) sparse gathered attention
    attn_sparse<<<dim3(S_F, NH), ATT_WAVES * 32, 0, stream>>>(
        q_ws, k_ws, v_ws, gix, ab, attn_o);
    // 4) output projection
    dim3 g2(NTOK / 128, CDIM / 128);
    gemm_f32_wmma<<<g2, 256, 0, stream>>>(attn_o, Wp, bp, out,
                                          NTOK, CDIM, CDIM);
}